// TrainableNonlinearity_25013889531939
// MI455X (gfx1250) — compile-verified
//
#include <hip/hip_runtime.h>

// ---------------------------------------------------------------------------
// Per-channel MLP 1->8->8->1 with ELU, fused into a single kernel.
//
// Layer 2 (the 64-MAC/element hot spot) runs on the CDNA5 matrix pipe via
// V_WMMA_F32_16X16X4_F32: two channels are packed block-diagonally into a
// 16x16 A matrix, 16 timesteps into B, and four chained K=4 WMMAs produce
// h2 for (2 channels x 16 t) per wave-iteration, fully in fp32.
//
// C/D layout (ISA 7.12.2): VGPR r holds row m=r (lanes 0-15, channel c0)
// or m=r+8 (lanes 16-31, channel c1); lane%16 = t. So after WMMA each lane
// owns the full 8-wide h2 vector of one (channel, t) -> layer 3 is a
// per-lane 8-FMA dot product, and the store is one float per lane.
//
// ELU = max(v,0) + (exp(min(v,0)) - 1): branchless AND VCC-free, so the 17
// independent ELU chains per tile can be interleaved/dual-issued and the
// v_exp_f32 TRANS hazard slots fill with neighboring chains' work (the
// previous cmp/cndmask form serialized every chain through vcc_lo).
// ---------------------------------------------------------------------------

typedef float v2f __attribute__((ext_vector_type(2)));
typedef float v8f __attribute__((ext_vector_type(8)));

#define NCH   128     // channels
#define WDIM  8       // hidden width
#define NB    8       // batch
#define NT    16000   // time
#define TTILE 16      // timesteps per WMMA tile
#define SUBW  8       // waves splitting the T axis per channel-pair
#define TILES_PER_WAVE ((NT / TTILE) / SUBW)   // 1000/8 = 125

__device__ __forceinline__ float eluf(float v) {
    // jax.nn.elu. Exact for v>0 since exp(0)==1. No compare, no VCC.
    const float e = __expf(fminf(v, 0.0f)) - 1.0f;
    return fmaxf(v, 0.0f) + e;
}

__global__ __launch_bounds__(256) void
TrainableNonlinearity_25013889531939_kernel(
    const float* __restrict__ x,   // [B, C, T]
    const float* __restrict__ w1,  // [C, W]
    const float* __restrict__ b1,  // [C, W]
    const float* __restrict__ w2,  // [C, W, W]  (out v, in w)
    const float* __restrict__ b2,  // [C, W]
    const float* __restrict__ w3,  // [C, W]
    const float* __restrict__ b3,  // [C]
    float* __restrict__ out)       // [B, C, T]
{
    const int lane   = threadIdx.x & 31;
    const int waveId = (blockIdx.x * blockDim.x + threadIdx.x) >> 5;

    const int totalWaves = NB * (NCH / 2) * SUBW;   // 4096
    if (waveId >= totalWaves) return;               // uniform per wave: EXEC stays all-1s

    const int pairIdx = waveId >> 3;                // / SUBW
    const int subWave = waveId & (SUBW - 1);
    const int b   = pairIdx >> 6;                   // / (NCH/2)
    const int cp  = pairIdx & 63;
    const int c0  = cp * 2;
    const int c1  = c0 + 1;

    const bool hi = (lane >= 16);
    const int  tl = lane & 15;                      // t within tile (both halves)
    const int  chLane = hi ? c1 : c0;               // output channel owned by this lane

    // ---- Hoist all weights into registers (amortized over 125 tiles) ----
    // A matrix (16x4 f32 layout, ISA 7.12.2): lane m = lane&15,
    // VGPR j of slice s holds K = 4*s + j + (hi ? 2 : 0).
    v2f   A[4];
    float w1v[4][2], b1v[4][2];
#pragma unroll
    for (int s = 0; s < 4; ++s) {
        const int cs = (s < 2) ? c0 : c1;           // K-slice's channel
        const int mch = (tl < 8) ? c0 : c1;         // output row's channel
#pragma unroll
        for (int j = 0; j < 2; ++j) {
            const int k  = 4 * s + j + (hi ? 2 : 0);
            const int kw = k & 7;                   // input index w within channel
            // block-diagonal A: zero when row channel != column channel
            A[s][j] = (mch == cs) ? w2[cs * (WDIM * WDIM) + (tl & 7) * WDIM + kw]
                                  : 0.0f;
            // B-side (layer 1) coefficients for the same (s, j, half) K index
            w1v[s][j] = w1[cs * WDIM + kw];
            b1v[s][j] = b1[cs * WDIM + kw];
        }
    }
    float b2v[WDIM], w3v[WDIM];
#pragma unroll
    for (int r = 0; r < WDIM; ++r) {
        b2v[r] = b2[chLane * WDIM + r];
        w3v[r] = w3[chLane * WDIM + r];
    }
    const float b3v = b3[chLane];

    const long xBase0 = (long)(b * NCH + c0) * NT;
    const long xBase1 = (long)(b * NCH + c1) * NT;
    const long oBase  = (long)(b * NCH + chLane) * NT;
    const int  tStart = subWave * TILES_PER_WAVE * TTILE;

    for (int tile = 0; tile < TILES_PER_WAVE; ++tile) {
        const int t0 = tStart + tile * TTILE;

        const float xv0 = x[xBase0 + t0 + tl];      // c0 sample at t
        const float xv1 = x[xBase1 + t0 + tl];      // c1 sample at t

        if (tile + 1 < TILES_PER_WAVE) {            // gfx1250 global_prefetch_b8
            __builtin_prefetch(&x[xBase0 + t0 + TTILE + tl], 0, 3);
            __builtin_prefetch(&x[xBase1 + t0 + TTILE + tl], 0, 3);
        }

        // ---- layer 1 (build B slices) + layer 2 (4 chained f32 WMMAs) ----
        v8f acc = {};
#pragma unroll
        for (int s = 0; s < 4; ++s) {
            const float xs = (s < 2) ? xv0 : xv1;
            v2f Bm;
            Bm.x = eluf(fmaf(xs, w1v[s][0], b1v[s][0]));
            Bm.y = eluf(fmaf(xs, w1v[s][1], b1v[s][1]));
            // (neg_a, A, neg_b, B, c_mod, C, reuse_a, reuse_b)
            acc = __builtin_amdgcn_wmma_f32_16x16x4_f32(
                false, A[s], false, Bm, (short)0, acc, false, false);
        }

        // ---- layer 2 bias+ELU, layer 3 dot+bias+ELU, store ----
        float dot = b3v;
#pragma unroll
        for (int r = 0; r < WDIM; ++r) {
            const float h2 = eluf(acc[r] + b2v[r]);
            dot = fmaf(h2, w3v[r], dot);
        }
        out[oBase + t0 + tl] = eluf(dot);
    }
}

extern "C" void kernel_launch(void* const* d_in, const int* in_sizes, int n_in,
                              void* d_out, int out_size, void* d_ws, size_t ws_size,
                              hipStream_t stream) {
    (void)in_sizes; (void)n_in; (void)out_size; (void)d_ws; (void)ws_size;
    const float* x  = (const float*)d_in[0];
    const float* w1 = (const float*)d_in[1];
    const float* b1 = (const float*)d_in[2];
    const float* w2 = (const float*)d_in[3];
    const float* b2 = (const float*)d_in[4];
    const float* w3 = (const float*)d_in[5];
    const float* b3 = (const float*)d_in[6];
    float* out = (float*)d_out;

    // 4096 waves = 8 batches x 64 channel-pairs x 8 T-slices
    const int threads = 256;                         // 8 wave32 per block
    const int totalThreads = NB * (NCH / 2) * SUBW * 32;
    const int blocks = totalThreads / threads;       // 512
    TrainableNonlinearity_25013889531939_kernel<<<blocks, threads, 0, stream>>>(
        x, w1, b1, w2, b2, w3, b3, out);
}